// BPCAUnpooling_79250736546211
// MI455X (gfx1250) — compile-verified
//
#include <hip/hip_runtime.h>
#include <hip/hip_bf16.h>

// BPCA unpooling for MI455X (gfx1250, wave32).
// X: [32, 65536, 16] fp32. Out: [32, 65536, 16] fp32 (same flat layout).
// Pass 1: G = X^T X via WMMA f32 16x16x4 (A==B trick), TDM double-buffered
//         LDS staging (tensor_load_to_lds, TENSORcnt).
// Pass 2: 16x16 Jacobi eigensolve per batch (32 single-wave blocks).
// Pass 3: orig = X @ comp via chained WMMA, *std+mean, NaN->0; TDM load in /
//         TDM store out with independent per-wave TENSORcnt counters.

typedef __attribute__((ext_vector_type(2))) float v2f;
typedef __attribute__((ext_vector_type(4))) float v4f;
typedef __attribute__((ext_vector_type(8))) float v8f;
typedef __attribute__((ext_vector_type(4))) unsigned int u32x4;
typedef __attribute__((ext_vector_type(8))) unsigned int u32x8;

#define NBATCH 32
#define NROWS  65536
#define NCH    16
#define GS_STRIDE 272   // 256 Gram + 16 column sums per batch

#define TILE_ROWS  256
#define TILE_ELEMS (TILE_ROWS * NCH)   // 4096 fp32 = 16 KB

// --------------------------------------------------------------- TDM issue --
// Builds a 2-group Tensor DMA Descriptor (1-D contiguous tile of `nelem`
// fp32 elements) in SGPRs and issues tensor_load_to_lds /
// tensor_store_from_lds. Tracked with TENSORcnt (per wave, in order).
__device__ __forceinline__ unsigned lds_byte_addr(const void* p) {
  // LDS aperture: generic addr[31:0] is the LDS byte offset.
  return (unsigned)(unsigned long long)(uintptr_t)p;
}

__device__ __forceinline__ void tdm_issue(const void* gptr, unsigned lds_addr,
                                          unsigned nelem, bool is_store) {
  unsigned long long ga = (unsigned long long)(uintptr_t)gptr;
  u32x4 g0;
  g0[0] = 1u;                                    // count=1, user descriptor
  g0[1] = lds_addr;                              // lds_addr [63:32]
  g0[2] = (unsigned)ga;                          // global_addr [95:64]
  g0[3] = (unsigned)((ga >> 32) & 0x01ffffffu)   // global_addr [120:96]
        | (2u << 30);                            // type=2 ("image") [127:126]
  u32x8 g1;
  g1[0] = 2u << 16;                              // data_size=2 (4B), mask=0
  g1[1] = (nelem & 0xffffu) << 16;               // tensor_dim0 lo16
  g1[2] = (nelem >> 16) | (1u << 16);            // tensor_dim0 hi16, dim1=1
  g1[3] = (nelem & 0xffffu) << 16;               // dim1 hi=0, tile_dim0=nelem
  g1[4] = 0u;                                    // tile_dim1=0 (1-D), dim2=0
  g1[5] = nelem;                                 // tensor_dim0_stride lo32
  g1[6] = (nelem & 0xffffu) << 16;               // d0s hi16=0, d1s lo16
  g1[7] = nelem >> 16;                           // tensor_dim1_stride hi32
  if (is_store)
    asm volatile("tensor_store_from_lds %0, %1" :: "s"(g0), "s"(g1) : "memory");
  else
    asm volatile("tensor_load_to_lds %0, %1" :: "s"(g0), "s"(g1) : "memory");
}

// ---------------------------------------------------------------- zero ws ---
__global__ void bpca_zero_ws(float* __restrict__ gs) {
  int i = blockIdx.x * blockDim.x + threadIdx.x;
  if (i < NBATCH * GS_STRIDE) gs[i] = 0.0f;
}

// ------------------------------------------------------------- pass 1: Gram -
// grid (8, 32), block 256 (8 waves). Block: 8192 rows = 32 tiles of 256.
// Wave 0 runs the TDM double-buffer pipeline; all waves compute from LDS.
// fp32 A-fragment (ISA 7.12.2): lane L holds M=L%16, K = vgpr + 2*(L/16);
// identical registers serve as the B fragment of the transposed chunk.
__global__ void bpca_gram_kernel(const float* __restrict__ x,
                                 float* __restrict__ gs) {
  __shared__ float buf[2][TILE_ELEMS];  // 2 x 16 KB staging
  __shared__ float red[GS_STRIDE];

  const int b    = blockIdx.y;
  const int wave = threadIdx.x >> 5;
  const int lane = threadIdx.x & 31;
  const int half = lane >> 4;
  const int c    = lane & 15;

  for (int i = threadIdx.x; i < GS_STRIDE; i += blockDim.x) red[i] = 0.0f;

  const int tiles = 8192 / TILE_ROWS;  // 32
  const float* gx = x + ((long)b * NROWS + (long)blockIdx.x * 8192) * NCH;
  const unsigned lbuf[2] = { lds_byte_addr(&buf[0][0]),
                             lds_byte_addr(&buf[1][0]) };

  if (wave == 0) tdm_issue(gx, lbuf[0], TILE_ELEMS, false);

  v8f acc = {};
  float sum = 0.0f;
  for (int t = 0; t < tiles; ++t) {
    if (wave == 0) {
      if (t + 1 < tiles) {
        tdm_issue(gx + (long)(t + 1) * TILE_ELEMS, lbuf[(t + 1) & 1],
                  TILE_ELEMS, false);
        __builtin_amdgcn_s_wait_tensorcnt((short)1);  // tile t arrived
      } else {
        __builtin_amdgcn_s_wait_tensorcnt((short)0);  // last tile arrived
      }
    }
    __syncthreads();  // tile t resident in buf[t&1]

    const float* tl = buf[t & 1] + wave * (TILE_ROWS / 8) * NCH;  // 32 rows
    for (int r = 0; r < TILE_ROWS / 8; r += 4) {
      float a0 = tl[(r + 2 * half + 0) * NCH + c];
      float a1 = tl[(r + 2 * half + 1) * NCH + c];
      v2f a;
      a.x = a0;
      a.y = a1;
      acc = __builtin_amdgcn_wmma_f32_16x16x4_f32(
          false, a, false, a, (short)0, acc, false, false);
      sum += a0 + a1;
    }
    __syncthreads();  // reads done before buf[t&1] is DMA-overwritten
  }

  // halves cover complementary row sets; fold
  sum += __shfl_xor(sum, 16);

  // C/D layout: VGPR r, lane L -> M = r + 8*(L/16), N = L%16
#pragma unroll
  for (int r = 0; r < 8; ++r) {
    int m = r + 8 * half;
    atomicAdd(&red[m * 16 + c], acc[r]);
  }
  if (half == 0) atomicAdd(&red[256 + c], sum);
  __syncthreads();

  float* g = gs + b * GS_STRIDE;
  for (int i = threadIdx.x; i < GS_STRIDE; i += blockDim.x)
    atomicAdd(&g[i], red[i]);
}

// --------------------------------------------- pass 2: Jacobi eig + stats ---
// 32 blocks x 32 threads (single wave: barriers are S_NOP; program order
// within the wave gives correctness).
__global__ void bpca_eig_kernel(const float* __restrict__ gs,
                                float* __restrict__ compT,
                                float* __restrict__ sb) {
  const int b   = blockIdx.x;
  const int tid = threadIdx.x;

  __shared__ float A[16][16];
  __shared__ float G[16][16];
  __shared__ float V[16][16];
  __shared__ float csum[16];
  __shared__ int   perm[16];

  const float* g = gs + b * GS_STRIDE;
  for (int i = tid; i < 256; i += 32) {
    float v = g[i];
    A[i >> 4][i & 15] = v;
    G[i >> 4][i & 15] = v;
    V[i >> 4][i & 15] = ((i >> 4) == (i & 15)) ? 1.0f : 0.0f;
  }
  if (tid < 16) csum[tid] = g[256 + tid] * (1.0f / (float)NROWS);
  __syncthreads();

  for (int sweep = 0; sweep < 8; ++sweep) {
    for (int p = 0; p < 15; ++p) {
      for (int q = p + 1; q < 16; ++q) {
        float apq = A[p][q];
        float app = A[p][p];
        float aqq = A[q][q];
        float cth = 1.0f, sth = 0.0f;
        if (fabsf(apq) > 1e-12f * (fabsf(app) + fabsf(aqq)) + 1e-30f) {
          float theta = (aqq - app) / (2.0f * apq);
          float t = 1.0f / (fabsf(theta) + sqrtf(theta * theta + 1.0f));
          if (theta < 0.0f) t = -t;
          cth = 1.0f / sqrtf(t * t + 1.0f);
          sth = t * cth;
        }
        __syncthreads();
        if (tid < 16) {  // A <- A*J, V <- V*J
          float aip = A[tid][p], aiq = A[tid][q];
          A[tid][p] = cth * aip - sth * aiq;
          A[tid][q] = sth * aip + cth * aiq;
          float vip = V[tid][p], viq = V[tid][q];
          V[tid][p] = cth * vip - sth * viq;
          V[tid][q] = sth * vip + cth * viq;
        }
        __syncthreads();
        if (tid < 16) {  // A <- J^T*A
          float api = A[p][tid], aqi = A[q][tid];
          A[p][tid] = cth * api - sth * aqi;
          A[q][tid] = sth * api + cth * aqi;
        }
        __syncthreads();
      }
    }
  }

  if (tid == 0) {  // sort eigenvalues descending (svd convention)
    for (int i = 0; i < 16; ++i) perm[i] = i;
    for (int i = 0; i < 15; ++i) {
      int best = i;
      for (int j = i + 1; j < 16; ++j)
        if (A[perm[j]][perm[j]] > A[perm[best]][perm[best]]) best = j;
      int t = perm[i]; perm[i] = perm[best]; perm[best] = t;
    }
  }
  __syncthreads();

  // comp[k][j] = vh[k][j] = V[j][perm[k]]; store transposed per column j
  float* ct = compT + b * 256;
  for (int i = tid; i < 256; i += 32) {
    int j = i >> 4, k = i & 15;
    ct[i] = V[j][perm[k]];
  }

  if (tid < 16) {  // per-column mean / std from csum and G
    int j = tid;
    float mean = 0.0f, e2 = 0.0f;
    for (int k = 0; k < 16; ++k) {
      float cjk = V[j][perm[k]];
      mean += csum[k] * cjk;
      float dot = 0.0f;
      for (int l = 0; l < 16; ++l) dot += G[k][l] * V[j][perm[l]];
      e2 += cjk * dot;
    }
    e2 *= (1.0f / (float)NROWS);
    float var = e2 - mean * mean;
    sb[b * 32 + j]      = sqrtf(fmaxf(var, 0.0f));  // std (ddof=0)
    sb[b * 32 + 16 + j] = mean;
  }
}

// ------------------------------------------------------ pass 3: projection --
// grid (16, 32), block 256 (8 waves). Block: 4096 rows = 16 tiles of 256.
// Wave 0 owns TDM loads, wave 7 owns TDM stores (separate per-wave
// TENSORcnt counters). X re-read hits the 192 MB L2 (X is 128 MB).
#define PT_TILES 16

__global__ void bpca_project_kernel(const float* __restrict__ x,
                                    const float* __restrict__ compT,
                                    const float* __restrict__ sb,
                                    float* __restrict__ out) {
  __shared__ float inb[2][TILE_ELEMS];   // 2 x 16 KB in
  __shared__ float oub[2][TILE_ELEMS];   // 2 x 16 KB out

  const int b    = blockIdx.y;
  const int wave = threadIdx.x >> 5;
  const int lane = threadIdx.x & 31;
  const int half = lane >> 4;
  const int c    = lane & 15;

  const long row0 = (long)blockIdx.x * (TILE_ROWS * PT_TILES);
  const float* gx = x + ((long)b * NROWS + row0) * NCH;
  float*       gy = out + ((long)b * NROWS + row0) * NCH;

  // B fragments: B[k][n] = comp[k][n]; lane holds k = 4*s + v + 2*half, n = c
  const float* ct = compT + b * 256 + c * 16;
  v2f bm[4];
#pragma unroll
  for (int s = 0; s < 4; ++s) {
    bm[s].x = ct[4 * s + 2 * half + 0];
    bm[s].y = ct[4 * s + 2 * half + 1];
  }
  const float scl  = sb[b * 32 + c];
  const float bias = sb[b * 32 + 16 + c];

  const unsigned lin[2] = { lds_byte_addr(&inb[0][0]),
                            lds_byte_addr(&inb[1][0]) };
  const unsigned lou[2] = { lds_byte_addr(&oub[0][0]),
                            lds_byte_addr(&oub[1][0]) };

  if (wave == 0) tdm_issue(gx, lin[0], TILE_ELEMS, false);

  for (int t = 0; t < PT_TILES; ++t) {
    if (wave == 0) {  // load pipeline
      if (t + 1 < PT_TILES) {
        tdm_issue(gx + (long)(t + 1) * TILE_ELEMS, lin[(t + 1) & 1],
                  TILE_ELEMS, false);
        __builtin_amdgcn_s_wait_tensorcnt((short)1);  // tile t arrived
      } else {
        __builtin_amdgcn_s_wait_tensorcnt((short)0);  // last tile arrived
      }
    }
    if (wave == 7 && t >= 2)  // store(t-2) must be done before reusing oub
      __builtin_amdgcn_s_wait_tensorcnt((short)1);
    __syncthreads();  // in-tile resident, out-tile free

    const float* ti = inb[t & 1] + wave * (TILE_ROWS / 8) * NCH;  // 32 rows
    float*       to = oub[t & 1] + wave * (TILE_ROWS / 8) * NCH;
#pragma unroll
    for (int s16 = 0; s16 < 2; ++s16) {  // two 16x16 subtiles per wave
      const float* tl = ti + s16 * 16 * NCH;
      v8f acc = {};
#pragma unroll
      for (int s = 0; s < 4; ++s) {  // K=16 as 4 chained K=4 WMMAs
        v2f a;
        a.x = tl[c * 16 + 4 * s + 2 * half + 0];
        a.y = tl[c * 16 + 4 * s + 2 * half + 1];
        acc = __builtin_amdgcn_wmma_f32_16x16x4_f32(
            false, a, false, bm[s], (short)0, acc, false, false);
      }
      float* td = to + s16 * 16 * NCH;
#pragma unroll
      for (int r = 0; r < 8; ++r) {
        int m = r + 8 * half;  // D row (sample), column = c
        float v = acc[r] * scl + bias;
        if (__builtin_isnan(v)) v = 0.0f;
        td[m * 16 + c] = v;
      }
    }
    __syncthreads();  // all LDS writes of out-tile visible
    if (wave == 7)
      tdm_issue(gy + (long)t * TILE_ELEMS, lou[t & 1], TILE_ELEMS, true);
  }
  if (wave == 7) __builtin_amdgcn_s_wait_tensorcnt((short)0);
}

// ---------------------------------------------------------------- launcher --
extern "C" void kernel_launch(void* const* d_in, const int* in_sizes, int n_in,
                              void* d_out, int out_size, void* d_ws, size_t ws_size,
                              hipStream_t stream) {
  const float* x = (const float*)d_in[0];
  float* out = (float*)d_out;

  // workspace layout (fp32): [32*272 Gram+sums][32*256 compT][32*32 std/mean]
  float* gs    = (float*)d_ws;
  float* compT = gs + NBATCH * GS_STRIDE;
  float* sb    = compT + NBATCH * 256;

  bpca_zero_ws<<<(NBATCH * GS_STRIDE + 255) / 256, 256, 0, stream>>>(gs);
  bpca_gram_kernel<<<dim3(8, NBATCH), 256, 0, stream>>>(x, gs);
  bpca_eig_kernel<<<NBATCH, 32, 0, stream>>>(gs, compT, sb);
  bpca_project_kernel<<<dim3(16, NBATCH), 256, 0, stream>>>(x, compT, sb, out);
}